// FCOSPostProcessor_50233937494364
// MI455X (gfx1250) — compile-verified
//
#include <hip/hip_runtime.h>
#include <stdint.h>

#ifndef FCOS_USE_ASYNC
#define FCOS_USE_ASYNC 1   // gfx1250 async global->LDS path (ASYNCcnt-tracked)
#endif

namespace {

constexpr int   kH = 400, kW = 608, kHW = kH * kW;   // fixed by reference harness
constexpr int   kSlot  = 1024;                       // selection slots (>= PRE_NMS_TOP_N)
constexpr int   kTopN  = 1000;                       // PRE_NMS_TOP_N
constexpr int   kPostN = 100;                        // FPN_POST_NMS_TOP_N
constexpr float kNmsTh = 0.6f;
constexpr float kClsLogitTh = -2.9444389791664403f;  // logit(0.05)
constexpr int   kHBits = 15;
constexpr int   kHBins = 1 << kHBits;                // 32768 bins = 128 KB LDS
constexpr int   kSub   = 64;                         // streaming blocks per image
constexpr int   kChunk = kHW / kSub;                 // 3800 elements (div by 4)
constexpr int   kTieCap = 4096;                      // deterministic tie buffer

typedef float vfloat4 __attribute__((ext_vector_type(4)));   // clang-native vec4

__device__ __forceinline__ float sigmoidf(float x) {
  return 1.0f / (1.0f + __expf(-x));
}
__device__ __forceinline__ unsigned f2s(float f) {   // monotone float->u32
  unsigned u = __float_as_uint(f);
  return (u & 0x80000000u) ? ~u : (u | 0x80000000u);
}
__device__ __forceinline__ float s2f(unsigned k) {
  return (k & 0x80000000u) ? __uint_as_float(k & 0x7FFFFFFFu)
                           : __uint_as_float(~k);
}

#if FCOS_USE_ASYNC
// Per-lane copies global -> LDS, tracked on ASYNCcnt (gfx1250).
__device__ __forceinline__ void async_b32(unsigned ldsOff, unsigned long long ga) {
  asm volatile("global_load_async_to_lds_b32 %0, %1, off" :: "v"(ldsOff), "v"(ga) : "memory");
}
__device__ __forceinline__ void async_b64(unsigned ldsOff, unsigned long long ga) {
  asm volatile("global_load_async_to_lds_b64 %0, %1, off" :: "v"(ldsOff), "v"(ga) : "memory");
}
__device__ __forceinline__ void async_b128(unsigned ldsOff, unsigned long long ga) {
  asm volatile("global_load_async_to_lds_b128 %0, %1, off" :: "v"(ldsOff), "v"(ga) : "memory");
}
__device__ __forceinline__ void wait_asynccnt0() {
  asm volatile("s_wait_asynccnt 0x0" ::: "memory");
}
__device__ __forceinline__ unsigned lds_off(const void* p) {
  return (unsigned)(size_t)p;            // generic LDS pointer: low 32 bits = LDS addr
}
__device__ __forceinline__ unsigned long long gaddr(const void* p) {
  return (unsigned long long)(size_t)p;
}
#endif

// ---------- shared phase helpers (used by split kernels and fallback) -------

// Find the histogram bin containing the kTopN-th largest key. hist/cs in LDS.
__device__ __forceinline__ void find_threshold(const unsigned* hist, unsigned* cs,
                                               int* sThrBin, int* sCntAbove, int t) {
  if (t == 0) { *sThrBin = -1; *sCntAbove = 0; }     // -1 => fewer than kTopN total
  const int rbase = t * 32;                          // 32 bins each, DESCENDING order
  unsigned c0 = 0;
#pragma unroll 4
  for (int u = 0; u < 32; ++u) c0 += hist[kHBins - 1 - (rbase + u)];
  cs[t] = c0;
  __syncthreads();
  for (int off = 1; off < 1024; off <<= 1) {         // Hillis-Steele inclusive scan
    unsigned v = (t >= off) ? cs[t - off] : 0u;
    __syncthreads();
    cs[t] += v;
    __syncthreads();
  }
  unsigned c = cs[t] - c0;                           // exclusive prefix
  for (int u = 0; u < 32; ++u) {
    const int bin = kHBins - 1 - (rbase + u);
    const unsigned h = hist[bin];
    if (c < (unsigned)kTopN && c + h >= (unsigned)kTopN) {  // unique crossing
      *sThrBin = bin; *sCntAbove = (int)c;
    }
    c += h;
  }
  __syncthreads();
}

// Bitonic sort of kSlot u64 keys in LDS, descending; one element per thread.
__device__ __forceinline__ void sort_keys_desc(unsigned long long* key, int t) {
  for (int kk = 2; kk <= kSlot; kk <<= 1) {
    for (int j = kk >> 1; j > 0; j >>= 1) {
      __syncthreads();
      const int ixj = t ^ j;
      if (ixj > t) {
        const unsigned long long a = key[t], b = key[ixj];
        const bool desc = ((t & kk) == 0);
        if (desc ? (a < b) : (a > b)) { key[t] = b; key[ixj] = a; }
      }
    }
  }
  __syncthreads();
}

// Bitonic sort of kTieCap u64 keys in LDS, descending; 1024 threads, 2 pairs each.
__device__ __forceinline__ void sort_ties_desc(unsigned long long* v, int t) {
  for (int kk = 2; kk <= kTieCap; kk <<= 1) {
    for (int j = kk >> 1; j > 0; j >>= 1) {
      __syncthreads();
      for (int p = t; p < kTieCap / 2; p += 1024) {
        const int i   = ((p & ~(j - 1)) << 1) | (p & (j - 1));
        const int ixj = i | j;
        const bool desc = ((i & kk) == 0);
        const unsigned long long a = v[i], b = v[ixj];
        if (desc ? (a < b) : (a > b)) { v[i] = b; v[ixj] = a; }
      }
    }
  }
  __syncthreads();
}

// Phases: decode+clip boxes (async gather of reg planes), build 1024x1024-bit
// suppression matrix, single-wave32 greedy keep scan.
__device__ __forceinline__ void decode_and_nms(const unsigned long long* key,
                                               float (*bx)[4],
                                               unsigned (*supp)[32],
                                               unsigned* keepOut,
                                               const float* regN,
                                               const int* imsz,
                                               int n, int t) {
  for (int i = t; i < kSlot * 32; i += 1024) (&supp[0][0])[i] = 0u;

  const unsigned l = (unsigned)key[t];               // flat location (0 if invalid)
#if FCOS_USE_ASYNC
#pragma unroll
  for (int c = 0; c < 4; ++c)                        // per-lane scattered gather
    async_b32(lds_off(&bx[t][c]), gaddr(regN + (size_t)c * kHW + l));
  wait_asynccnt0();
#else
#pragma unroll
  for (int c = 0; c < 4; ++c) bx[t][c] = regN[(size_t)c * kHW + l];
#endif
  const float px = (float)(l % kW) * 8.0f + 4.0f;    // stride-8 FCOS grid
  const float py = (float)(l / kW) * 8.0f + 4.0f;
  const float wmax = (float)imsz[2 * n + 1] - 1.0f;
  const float hmax = (float)imsz[2 * n + 0] - 1.0f;
  const float dl = bx[t][0], dtp = bx[t][1], dr = bx[t][2], dbt = bx[t][3];
  const float x1 = fminf(fmaxf(px - dl, 0.0f), wmax);
  const float y1 = fminf(fmaxf(py - dtp, 0.0f), hmax);
  const float x2 = fminf(fmaxf(px + dr, 0.0f), wmax);
  const float y2 = fminf(fmaxf(py + dbt, 0.0f), hmax);
  bx[t][0] = x1; bx[t][1] = y1; bx[t][2] = x2; bx[t][3] = y2;
  const float areaJ = (x2 - x1) * (y2 - y1);
  __syncthreads();                                   // boxes + zeroed supp visible

  for (int i = 0; i < t; ++i) {                      // column j=t vs rows i<t
    const float bx1 = bx[i][0], by1 = bx[i][1], bx2 = bx[i][2], by2 = bx[i][3];
    const float areaI = (bx2 - bx1) * (by2 - by1);
    const float iw = fmaxf(fminf(x2, bx2) - fmaxf(x1, bx1), 0.0f);
    const float ih = fmaxf(fminf(y2, by2) - fmaxf(y1, by1), 0.0f);
    const float inter = iw * ih;
    const float iou = inter / (areaI + areaJ - inter + 1e-9f);
    if (iou > kNmsTh)
      atomicOr(&supp[i][t >> 5], 1u << (t & 31));    // sparse: real overlaps only
  }
  __syncthreads();

  if (t < 32) {                                      // greedy scan: one wave32, no barriers
    unsigned kw = 0xFFFFFFFFu;
    for (int i = 0; i < kSlot; ++i) {
      if ((unsigned)(key[i] >> 32) == 0u) break;     // sorted: rest invalid
      const unsigned w = __shfl(kw, i >> 5, 32);
      if ((w >> (i & 31)) & 1u) kw &= ~supp[i][t];   // 32 lanes -> 32 banks
    }
    keepOut[t] = kw;
  }
  __syncthreads();
}

__device__ __forceinline__ void emit_results(const unsigned long long* key,
                                             const float (*bx)[4],
                                             const unsigned* keepOut,
                                             float* out, int n, int N, int t) {
  if (t != 0) return;
  float* dets   = out + (size_t)n * kPostN * 5;
  float* labels = out + (size_t)N * kPostN * 5 + (size_t)n * kPostN;
  float* valid  = out + (size_t)N * kPostN * 5 + (size_t)N * kPostN + (size_t)n * kPostN;
  int cnt = 0;
  for (int i = 0; i < kSlot && cnt < kPostN; ++i) {
    const unsigned sk = (unsigned)(key[i] >> 32);
    if (sk == 0u) break;
    if ((keepOut[i >> 5] >> (i & 31)) & 1u) {
      dets[cnt * 5 + 0] = bx[i][0]; dets[cnt * 5 + 1] = bx[i][1];
      dets[cnt * 5 + 2] = bx[i][2]; dets[cnt * 5 + 3] = bx[i][3];
      dets[cnt * 5 + 4] = s2f(sk);
      labels[cnt] = 1.0f; valid[cnt] = 1.0f; ++cnt;
    }
  }
  for (; cnt < kPostN; ++cnt) {
    dets[cnt * 5 + 0] = 0.0f; dets[cnt * 5 + 1] = 0.0f; dets[cnt * 5 + 2] = 0.0f;
    dets[cnt * 5 + 3] = 0.0f; dets[cnt * 5 + 4] = 0.0f;
    labels[cnt] = 0.0f; valid[cnt] = 0.0f;
  }
}

} // namespace

// ======================= split pipeline (HBM-saturating) =====================

extern "C" __global__ void fcos_zero_u32(unsigned* p, long n) {
  long i = (long)blockIdx.x * blockDim.x + threadIdx.x;
  const long stride = (long)gridDim.x * blockDim.x;
  for (; i < n; i += stride) p[i] = 0u;
}

// K1: 64 blocks/image x 256 threads (~8K waves total -> HBM-rate streaming).
// Streams cls+ctr once (nontemporal b128), emits 4B sortable key plane + histogram.
extern "C" __global__ void __launch_bounds__(256)
fcos_k1_score(const float* __restrict__ cls, const float* __restrict__ ctr,
              unsigned* __restrict__ keys, unsigned* __restrict__ hist) {
  const int n = blockIdx.y, b = blockIdx.x, t = threadIdx.x;
  const size_t ibase = (size_t)n * kHW;
  const int cbase = b * kChunk;
  const float* clsN = cls + ibase;
  const float* ctrN = ctr + ibase;
  unsigned* keysN = keys + ibase;
  unsigned* histN = hist + (size_t)n * kHBins;
  for (int e = t * 4; e < kChunk; e += 1024) {
    const int i = cbase + e;
    __builtin_prefetch(clsN + i + 2048, 0, 0);
    __builtin_prefetch(ctrN + i + 2048, 0, 0);
    const vfloat4 c4 = __builtin_nontemporal_load((const vfloat4*)(clsN + i));
    const vfloat4 e4 = __builtin_nontemporal_load((const vfloat4*)(ctrN + i));
    unsigned kv[4];
#pragma unroll
    for (int u = 0; u < 4; ++u) {
      unsigned k32 = 0u;
      if (c4[u] > kClsLogitTh) {
        k32 = f2s(sigmoidf(c4[u]) * sigmoidf(e4[u]));   // MSB set => never 0
        atomicAdd(&histN[k32 >> (32 - kHBits)], 1u);
      }
      kv[u] = k32;
    }
    *(uint4*)(keysN + i) = make_uint4(kv[0], kv[1], kv[2], kv[3]);
  }
}

// K2: one block/image. Async-copies the 128 KB histogram into LDS, finds the
// kTopN threshold bin, writes {thrBin, cntAbove, need} params.
extern "C" __global__ void __launch_bounds__(1024)
fcos_k2_thresh(const unsigned* __restrict__ hist, int* __restrict__ params) {
  __shared__ unsigned lhist[kHBins];                 // 128 KB
  __shared__ unsigned cs[kSlot];
  __shared__ int sThr, sAbove;
  const int n = blockIdx.x, t = threadIdx.x;
  const unsigned* hN = hist + (size_t)n * kHBins;
#if FCOS_USE_ASYNC
#pragma unroll
  for (int u = 0; u < 8; ++u) {
    const int bin = u * 4096 + t * 4;                // coalesced 16B per lane
    async_b128(lds_off(&lhist[bin]), gaddr(hN + bin));
  }
  wait_asynccnt0();
#else
  for (int i = t; i < kHBins; i += 1024) lhist[i] = hN[i];
#endif
  __syncthreads();
  find_threshold(lhist, cs, &sThr, &sAbove, t);
  if (t == 0) {
    const int thr = sThr, ca = sAbove;
    params[n * 4 + 0] = thr;
    params[n * 4 + 1] = ca;
    params[n * 4 + 2] = (thr < 0) ? kSlot : (kTopN - ca);
  }
}

// K3: 64 blocks/image x 256. Reads the key plane (L2-resident), compacts the
// deterministic strictly-above set into slots; bin-equal ties go to a buffer.
extern "C" __global__ void __launch_bounds__(256)
fcos_k3_select(const unsigned* __restrict__ keys, const int* __restrict__ params,
               int* __restrict__ counters, unsigned long long* __restrict__ slots,
               unsigned long long* __restrict__ tiebuf) {
  const int n = blockIdx.y, b = blockIdx.x, t = threadIdx.x;
  const int thr = params[n * 4 + 0];
  const unsigned* keysN = keys + (size_t)n * kHW;
  unsigned long long* slotN = slots + (size_t)n * kSlot;
  unsigned long long* tieN  = tiebuf + (size_t)n * kTieCap;
  int* hi  = &counters[n * 2 + 0];
  int* tie = &counters[n * 2 + 1];
  const int cbase = b * kChunk;
  for (int e = t * 4; e < kChunk; e += 1024) {
    const int i = cbase + e;
    const uint4 kv4 = *(const uint4*)(keysN + i);
    const unsigned kv[4] = {kv4.x, kv4.y, kv4.z, kv4.w};
#pragma unroll
    for (int u = 0; u < 4; ++u) {
      const unsigned k32 = kv[u];
      if (!k32) continue;
      const int bin = (int)(k32 >> (32 - kHBits));
      const unsigned long long k64 =
          ((unsigned long long)k32 << 32) | (unsigned)(i + u);
      if (bin > thr) {                               // deterministic SET; sort fixes order
        const int s = atomicAdd(hi, 1);
        if (s < kSlot) slotN[s] = k64;
      } else if (bin == thr) {                       // ties: buffer, sort later
        const int tt = atomicAdd(tie, 1);
        if (tt < kTieCap) tieN[tt] = k64;
      }
    }
  }
}

// K4: one block/image. Sorts ties deterministically, fills slots, sorts all
// 1024 keys, then decode + bit-matrix NMS + emit.
extern "C" __global__ void __launch_bounds__(1024)
fcos_k4_nms(const unsigned long long* __restrict__ slots,
            const int* __restrict__ counters, const int* __restrict__ params,
            const unsigned long long* __restrict__ tiebuf,
            const float* __restrict__ reg, const int* __restrict__ imsz,
            float* __restrict__ out, int N) {
  __shared__ union { unsigned long long ties[kTieCap]; unsigned supp[kSlot][32]; } u; // 128 KB
  __shared__ float bx[kSlot][4];
  __shared__ unsigned long long key[kSlot];
  __shared__ unsigned keepOut[32];
  const int n = blockIdx.x, t = threadIdx.x;
  const unsigned long long* slotN = slots + (size_t)n * kSlot;
  const unsigned long long* tieN  = tiebuf + (size_t)n * kTieCap;
  const int cAbove = params[n * 4 + 1];
  const int need   = params[n * 4 + 2];
  const int tieCnt = min(counters[n * 2 + 1], kTieCap);

#if FCOS_USE_ASYNC
  async_b64(lds_off(&key[t]), gaddr(slotN + t));
  wait_asynccnt0();
#else
  key[t] = slotN[t];
#endif
  for (int i = t; i < kTieCap; i += 1024)
    u.ties[i] = (i < tieCnt) ? tieN[i] : 0ull;       // 0 sorts last (descending)
  __syncthreads();
  sort_ties_desc(u.ties, t);                         // deterministic tie winners
  for (int r = t; r < need && r < tieCnt; r += 1024)
    key[cAbove + r] = u.ties[r];
  __syncthreads();

  sort_keys_desc(key, t);
  decode_and_nms(key, bx, u.supp, keepOut, reg + (size_t)n * 4 * kHW, imsz, n, t);
  emit_results(key, bx, keepOut, out, n, N, t);
}

// ================= fallback: single mega-kernel (tiny workspace) =============

extern "C" __global__ void __launch_bounds__(1024)
fcos_post_kernel(const float* __restrict__ cls, const float* __restrict__ reg,
                 const float* __restrict__ ctr, const int* __restrict__ imsz,
                 float* __restrict__ out, int N) {
  __shared__ union { unsigned hist[kHBins]; unsigned supp[kSlot][32]; } big; // 128 KB
  __shared__ union { unsigned cs[kSlot]; float bx[kSlot][4]; } scr;          //  16 KB
  __shared__ unsigned long long key[kSlot];
  __shared__ unsigned keepOut[32];
  __shared__ int sThrBin, sCntAbove, sNeed, sHi, sTie;

  const int t = threadIdx.x, n = blockIdx.x;
  const float* clsN = cls + (size_t)n * kHW;
  const float* ctrN = ctr + (size_t)n * kHW;

  for (int i = t; i < kHBins; i += 1024) big.hist[i] = 0u;
  if (t == 0) { sHi = 0; sTie = 0; }
  __syncthreads();

  for (int base = t * 4; base < kHW; base += 4096) {
    __builtin_prefetch(clsN + base + 16384, 0, 1);
    __builtin_prefetch(ctrN + base + 16384, 0, 1);
    const float4 c4 = *(const float4*)(clsN + base);
    const float4 e4 = *(const float4*)(ctrN + base);
    const float cv[4] = {c4.x, c4.y, c4.z, c4.w};
    const float ev[4] = {e4.x, e4.y, e4.z, e4.w};
#pragma unroll
    for (int uu = 0; uu < 4; ++uu)
      if (cv[uu] > kClsLogitTh)
        atomicAdd(&big.hist[f2s(sigmoidf(cv[uu]) * sigmoidf(ev[uu])) >> (32 - kHBits)], 1u);
  }
  __syncthreads();

  find_threshold(big.hist, scr.cs, &sThrBin, &sCntAbove, t);
  if (t == 0) sNeed = (sThrBin < 0) ? kSlot : (kTopN - sCntAbove);
  key[t] = 0ull;
  __syncthreads();

  {
    const int thr = sThrBin, cAbove = sCntAbove, need = sNeed;
    for (int base = t * 4; base < kHW; base += 4096) {
      const float4 c4 = *(const float4*)(clsN + base);
      const float4 e4 = *(const float4*)(ctrN + base);
      const float cv[4] = {c4.x, c4.y, c4.z, c4.w};
      const float ev[4] = {e4.x, e4.y, e4.z, e4.w};
#pragma unroll
      for (int uu = 0; uu < 4; ++uu) {
        if (cv[uu] > kClsLogitTh) {
          const unsigned k32 = f2s(sigmoidf(cv[uu]) * sigmoidf(ev[uu]));
          const int bin = (int)(k32 >> (32 - kHBits));
          int slot = -1;
          if (bin > thr) slot = atomicAdd(&sHi, 1);
          else if (bin == thr) {
            const int tt = atomicAdd(&sTie, 1);
            if (tt < need) slot = cAbove + tt;       // first-come ties (fallback only)
          }
          if (slot >= 0 && slot < kSlot)
            key[slot] = ((unsigned long long)k32 << 32) | (unsigned)(base + uu);
        }
      }
    }
  }
  __syncthreads();

  sort_keys_desc(key, t);
  decode_and_nms(key, scr.bx, big.supp, keepOut, reg + (size_t)n * 4 * kHW, imsz, n, t);
  emit_results(key, scr.bx, keepOut, out, n, N, t);
}

// ============================== launcher ====================================

extern "C" void kernel_launch(void* const* d_in, const int* in_sizes, int n_in,
                              void* d_out, int out_size, void* d_ws, size_t ws_size,
                              hipStream_t stream) {
  // inputs: locations, box_cls, box_regression, centerness, image_sizes
  const float* cls  = (const float*)d_in[1];
  const float* reg  = (const float*)d_in[2];
  const float* ctr  = (const float*)d_in[3];
  const int*   imsz = (const int*)d_in[4];
  const int N = in_sizes[1] / kHW;
  (void)n_in; (void)out_size;

  const size_t Bh = (size_t)N * kHBins * 4;          // per-image histograms
  const size_t Bs = (size_t)N * kSlot * 8;           // slot keys
  const size_t Bc = (size_t)N * 2 * 4;               // hi/tie counters
  const size_t Bp = (size_t)N * 4 * 4;               // params
  const size_t Bt = (size_t)N * kTieCap * 8;         // tie buffers
  const size_t Bk = (size_t)N * kHW * 4;             // sortable key plane
  const size_t need = Bh + Bs + Bc + Bp + Bt + Bk;

  if (d_ws && ws_size >= need) {
    char* w = (char*)d_ws;
    unsigned*           hist     = (unsigned*)(w);
    unsigned long long* slots    = (unsigned long long*)(w + Bh);
    int*                counters = (int*)(w + Bh + Bs);
    int*                params   = (int*)(w + Bh + Bs + Bc);
    unsigned long long* tiebuf   = (unsigned long long*)(w + Bh + Bs + Bc + Bp);
    unsigned*           keys     = (unsigned*)(w + Bh + Bs + Bc + Bp + Bt);

    fcos_zero_u32<<<1024, 256, 0, stream>>>(hist, (long)N * kHBins);
    fcos_zero_u32<<<256, 256, 0, stream>>>((unsigned*)slots, (long)N * (kSlot * 2 + 2));
    fcos_k1_score <<<dim3(kSub, N), 256, 0, stream>>>(cls, ctr, keys, hist);
    fcos_k2_thresh<<<N, 1024, 0, stream>>>(hist, params);
    fcos_k3_select<<<dim3(kSub, N), 256, 0, stream>>>(keys, params, counters, slots, tiebuf);
    fcos_k4_nms   <<<N, 1024, 0, stream>>>(slots, counters, params, tiebuf,
                                           reg, imsz, (float*)d_out, N);
  } else {
    fcos_post_kernel<<<N, 1024, 0, stream>>>(cls, reg, ctr, imsz, (float*)d_out, N);
  }
}